// QuantumLayer_86775519248977
// MI455X (gfx1250) — compile-verified
//
#include <hip/hip_runtime.h>
#include <hip/hip_bf16.h>

typedef __attribute__((ext_vector_type(2))) float v2f;
typedef __attribute__((ext_vector_type(8))) float v8f;

#define N_Q 12  // qubits / output columns (reference fixes N_QUBITS = 12)

// out[b,i] = prod_{k<=i} cos(x[b,k] + w[2k])   (closed form of the circuit)
// Computed as triangular matmul in log domain + exact sign-parity matmul,
// both on V_WMMA_F32_16X16X4_F32 (K chained 4+4+4 = 12).
__global__ __launch_bounds__(256) void qlayer_wmma_kernel(
    const float* __restrict__ X,   // [B, 12]
    const float* __restrict__ W,   // [72], only W[2k] used
    float* __restrict__ Out,       // [B, 12]
    int Brows, int ntiles) {
  const int lane = threadIdx.x & 31;
  const int waveInBlock = threadIdx.x >> 5;
  const int tile = blockIdx.x * (blockDim.x >> 5) + waveInBlock;
  if (tile >= ntiles) return;  // uniform per-wave: EXEC stays all-ones for WMMA

  const int half = lane >> 4;   // 0: lanes 0-15, 1: lanes 16-31
  const int l16  = lane & 15;
  const int koff = half * 2;    // A/B K-offset per lane half (ISA 16x4 f32 layout)

  // Each lane owns batch row (tile*16 + l16) for the A matrix, K = 4*kk+koff+{0,1}
  const int rowA = tile * 16 + l16;
  const int rowClamped = (rowA < Brows) ? rowA : (Brows - 1);

  v2f aG[3], aS[3], bU[3];
#pragma unroll
  for (int kk = 0; kk < 3; ++kk) {
#pragma unroll
    for (int j = 0; j < 2; ++j) {
      const int k = 4 * kk + koff + j;            // global K index, 0..11
      const float theta = X[rowClamped * N_Q + k] + W[2 * k];
      const float c = cosf(theta);
      // log-magnitude, clamped so 0-cosines give exp(sum)~0 instead of inf/NaN
      aG[kk][j] = fmaxf(logf(fabsf(c)), -80.0f);
      aS[kk][j] = (c < 0.0f) ? 1.0f : 0.0f;       // negativity indicator
      // B = upper-triangular ones: U[k, n] = (k <= n); lane%16 = column n
      bU[kk][j] = (k <= l16) ? 1.0f : 0.0f;
    }
  }

  v8f accG = {};  // sum_{k<=n} log|c_k|  per (row m, col n)
  v8f accS = {};  // sum_{k<=n} [c_k < 0] per (row m, col n)  (exact integers)
#pragma unroll
  for (int kk = 0; kk < 3; ++kk) {
    accG = __builtin_amdgcn_wmma_f32_16x16x4_f32(
        false, aG[kk], false, bU[kk], (short)0, accG, false, false);
  }
#pragma unroll
  for (int kk = 0; kk < 3; ++kk) {
    accS = __builtin_amdgcn_wmma_f32_16x16x4_f32(
        false, aS[kk], false, bU[kk], (short)0, accS, false, false);
  }

  // D layout: VGPR r -> row m = r + 8*half, column n = lane%16
  const int n = l16;
  if (n < N_Q) {
#pragma unroll
    for (int r = 0; r < 8; ++r) {
      const int m = r + half * 8;
      const int row = tile * 16 + m;
      if (row < Brows) {
        const int parity = __float2int_rn(accS[r]) & 1;
        const float mag = expf(accG[r]);
        Out[row * N_Q + n] = parity ? -mag : mag;
      }
    }
  }
}

extern "C" void kernel_launch(void* const* d_in, const int* in_sizes, int n_in,
                              void* d_out, int out_size, void* d_ws, size_t ws_size,
                              hipStream_t stream) {
  const float* X = (const float*)d_in[0];        // [B, 12] float32
  const float* W = (const float*)d_in[1];        // [72]    float32
  float* Out = (float*)d_out;                    // [B, 12] float32

  const int Brows = in_sizes[0] / N_Q;           // 2048
  const int ntiles = (Brows + 15) / 16;          // 128
  const int wavesPerBlock = 8;                   // 256 threads = 8 wave32
  const int blocks = (ntiles + wavesPerBlock - 1) / wavesPerBlock;

  qlayer_wmma_kernel<<<blocks, 32 * wavesPerBlock, 0, stream>>>(
      X, W, Out, Brows, ntiles);
}